// GIN_41351945126321
// MI455X (gfx1250) — compile-verified
//
#include <hip/hip_runtime.h>
#include <hip/hip_bf16.h>

#define N_NODES 50000
#define N_EDGES 800000
#define HIDDEN 128
#define N_LAYERS 3
#define N_GRAPHS 512
#define LN_EPS 1e-5f

typedef __attribute__((ext_vector_type(2))) float v2f;
typedef __attribute__((ext_vector_type(8))) float v8f;

#define LDSTRIDE 132   // 128 + 4 pad -> conflict-free strided LDS access
#define WSTRIDE 132    // padded stride for transposed weight tile
#define MTILE 64       // node rows per block (4 waves x 16-row WMMA tiles)
#define TB 128         // threads per block = 4 wave32

__global__ void zero_kernel(float* __restrict__ p, int n) {
  int i = blockIdx.x * blockDim.x + threadIdx.x;
  if (i < n) p[i] = 0.0f;
}

// x[n, :] = z_table[z[n], :]
__global__ void embed_kernel(const int* __restrict__ z,
                             const float* __restrict__ tab,
                             float* __restrict__ x) {
  int t = blockIdx.x * blockDim.x + threadIdx.x;
  int n = t >> 5, q = t & 31;
  if (n >= N_NODES) return;
  const float4 v = *(const float4*)(tab + z[n] * HIDDEN + q * 4);
  *(float4*)(x + n * HIDDEN + q * 4) = v;
}

__global__ void copy_kernel(const float4* __restrict__ s,
                            float4* __restrict__ d, int n4) {
  int i = blockIdx.x * blockDim.x + threadIdx.x;
  if (i < n4) d[i] = s[i];
}

// one wave per edge: h[dst] += x[src] (128 floats, 4 f32 atomics per lane)
__global__ void scatter_kernel(const float* __restrict__ x,
                               const int* __restrict__ src,
                               const int* __restrict__ dst,
                               float* __restrict__ h) {
  int t = blockIdx.x * blockDim.x + threadIdx.x;
  int e = t >> 5, q = t & 31;
  if (e >= N_EDGES) return;
  int s = src[e], d = dst[e];
  const float4 v = *(const float4*)(x + s * HIDDEN + q * 4);
  float* hp = h + d * HIDDEN + q * 4;
  atomicAdd(hp + 0, v.x);
  atomicAdd(hp + 1, v.y);
  atomicAdd(hp + 2, v.z);
  atomicAdd(hp + 3, v.w);
}

// xout = LayerNorm(relu(relu(hin @ W1 + B1) @ W2 + B2))   for 64 node rows
__global__ __launch_bounds__(TB) void mlp_kernel(
    const float* __restrict__ hin,
    const float* __restrict__ W1, const float* __restrict__ B1,
    const float* __restrict__ W2, const float* __restrict__ B2,
    const float* __restrict__ LG, const float* __restrict__ LB,
    float* __restrict__ xout) {
  __shared__ float sIn[MTILE * LDSTRIDE];
  __shared__ float sMid[MTILE * LDSTRIDE];
  __shared__ float sW[HIDDEN * WSTRIDE];   // W^T staged here (W1 then W2)

  const int t = threadIdx.x;
  const int lane = t & 31;
  const int wave = t >> 5;
  const int rowBase = blockIdx.x * MTILE;

  // cooperative load of the 64x128 input tile (zero-padded past N_NODES)
  for (int i = t; i < MTILE * (HIDDEN / 4); i += TB) {
    int r = i >> 5;
    int c4 = (i & 31) * 4;
    int n = rowBase + r;
    float4 v = make_float4(0.f, 0.f, 0.f, 0.f);
    if (n < N_NODES) v = *(const float4*)(hin + n * HIDDEN + c4);
    float* p = &sIn[r * LDSTRIDE + c4];
    p[0] = v.x; p[1] = v.y; p[2] = v.z; p[3] = v.w;
  }
  // stage sW = W1^T  (coalesced global read, transposed LDS store)
  for (int i = t; i < HIDDEN * HIDDEN; i += TB) {
    int k = i >> 7, c = i & 127;
    sW[c * WSTRIDE + k] = W1[i];
  }
  __syncthreads();

  const int rloc = lane & 15;   // A-row / B-col within 16-tile
  const int half = lane >> 4;   // K-half select (ISA 16x4 A layout)
  const int tileRow = wave * 16;

  // ---- GEMM1: sMid = relu(sIn @ W1 + B1), all-LDS inner loop ----
  for (int nt = 0; nt < 8; ++nt) {
    v8f c = {};
    const int col = nt * 16 + rloc;
#pragma unroll
    for (int kk = 0; kk < 32; ++kk) {
      const int kb = kk * 4 + half * 2;
      v2f a, b;
      const float* ap = &sIn[(tileRow + rloc) * LDSTRIDE + kb];
      a.x = ap[0]; a.y = ap[1];
      const float* bp = &sW[col * WSTRIDE + kb];
      b.x = bp[0]; b.y = bp[1];
      c = __builtin_amdgcn_wmma_f32_16x16x4_f32(false, a, false, b,
                                                (short)0, c, false, false);
    }
    const float bias = B1[col];
#pragma unroll
    for (int r = 0; r < 8; ++r) {
      float v = c[r] + bias;
      v = v > 0.f ? v : 0.f;
      sMid[(tileRow + r + half * 8) * LDSTRIDE + col] = v;
    }
  }
  __syncthreads();

  // restage sW = W2^T
  for (int i = t; i < HIDDEN * HIDDEN; i += TB) {
    int k = i >> 7, c = i & 127;
    sW[c * WSTRIDE + k] = W2[i];
  }
  __syncthreads();

  // ---- GEMM2: sIn = relu(sMid @ W2 + B2)  (reuse sIn) ----
  for (int nt = 0; nt < 8; ++nt) {
    v8f c = {};
    const int col = nt * 16 + rloc;
#pragma unroll
    for (int kk = 0; kk < 32; ++kk) {
      const int kb = kk * 4 + half * 2;
      v2f a, b;
      const float* ap = &sMid[(tileRow + rloc) * LDSTRIDE + kb];
      a.x = ap[0]; a.y = ap[1];
      const float* bp = &sW[col * WSTRIDE + kb];
      b.x = bp[0]; b.y = bp[1];
      c = __builtin_amdgcn_wmma_f32_16x16x4_f32(false, a, false, b,
                                                (short)0, c, false, false);
    }
    const float bias = B2[col];
#pragma unroll
    for (int r = 0; r < 8; ++r) {
      float v = c[r] + bias;
      v = v > 0.f ? v : 0.f;
      sIn[(tileRow + r + half * 8) * LDSTRIDE + col] = v;
    }
  }
  __syncthreads();

  // ---- LayerNorm: 2 lanes per row (64 cols each), shfl_xor(16) combine ----
  {
    const int r = lane & 15;
    const int cbase = half * 64;
    const float* rp = &sIn[(tileRow + r) * LDSTRIDE];
    float s = 0.f, sq = 0.f;
    for (int c = 0; c < 64; ++c) {
      float v = rp[cbase + c];
      s += v; sq += v * v;
    }
    s  += __shfl_xor(s, 16, 32);
    sq += __shfl_xor(sq, 16, 32);
    const float mean = s * (1.0f / HIDDEN);
    const float var  = sq * (1.0f / HIDDEN) - mean * mean;
    const float inv  = rsqrtf(var + LN_EPS);
    const int n = rowBase + tileRow + r;
    if (n < N_NODES) {
      for (int c = 0; c < 64; ++c) {
        const int cc = cbase + c;
        float v = (rp[cc] - mean) * inv * LG[cc] + LB[cc];
        xout[n * HIDDEN + cc] = v;
      }
    }
  }
}

// pool[batch[n], layer*H + d] += x[n, d]
__global__ void pool_kernel(const float* __restrict__ x,
                            const int* __restrict__ batch,
                            float* __restrict__ pool, int layer) {
  int t = blockIdx.x * blockDim.x + threadIdx.x;
  if (t >= N_NODES * HIDDEN) return;
  int n = t >> 7, d = t & 127;
  atomicAdd(&pool[batch[n] * (N_LAYERS * HIDDEN) + layer * HIDDEN + d], x[t]);
}

__global__ void cnt_kernel(const int* __restrict__ batch,
                           float* __restrict__ cnt) {
  int n = blockIdx.x * blockDim.x + threadIdx.x;
  if (n < N_NODES) atomicAdd(&cnt[batch[n]], 1.0f);
}

// out[g] = relu(pooled[g] @ lin1_W + lin1_b) @ lin2_W + lin2_b
__global__ __launch_bounds__(128) void head_kernel(
    const float* __restrict__ pool, const float* __restrict__ cnt,
    const float* __restrict__ W1, const float* __restrict__ B1,
    const float* __restrict__ W2, const float* __restrict__ B2,
    float* __restrict__ out) {
  const int g = blockIdx.x;
  const int t = threadIdx.x;
  __shared__ float pr[N_LAYERS * HIDDEN];
  __shared__ float r4[4];
  const float cgv = cnt[g];
  const float invc = 1.0f / (cgv > 1.0f ? cgv : 1.0f);
  for (int k = t; k < N_LAYERS * HIDDEN; k += 128)
    pr[k] = pool[g * (N_LAYERS * HIDDEN) + k] * invc;
  __syncthreads();
  float a = 0.f;
  for (int k = 0; k < N_LAYERS * HIDDEN; ++k)
    a += pr[k] * W1[k * HIDDEN + t];
  a += B1[t];
  a = a > 0.f ? a : 0.f;
  float p = a * W2[t];
  for (int off = 16; off; off >>= 1) p += __shfl_xor(p, off, 32);
  if ((t & 31) == 0) r4[t >> 5] = p;
  __syncthreads();
  if (t == 0) out[g] = r4[0] + r4[1] + r4[2] + r4[3] + B2[0];
}

extern "C" void kernel_launch(void* const* d_in, const int* in_sizes, int n_in,
                              void* d_out, int out_size, void* d_ws, size_t ws_size,
                              hipStream_t stream) {
  const int*   z     = (const int*)d_in[0];
  const int*   eidx  = (const int*)d_in[1];
  const int*   batch = (const int*)d_in[2];
  const float* ztab  = (const float*)d_in[3];
  const float* W1    = (const float*)d_in[4];
  const float* b1    = (const float*)d_in[5];
  const float* W2    = (const float*)d_in[6];
  const float* b2    = (const float*)d_in[7];
  const float* lng   = (const float*)d_in[8];
  const float* lnb   = (const float*)d_in[9];
  const float* l1W   = (const float*)d_in[10];
  const float* l1b   = (const float*)d_in[11];
  const float* l2W   = (const float*)d_in[12];
  const float* l2b   = (const float*)d_in[13];
  float* out = (float*)d_out;

  float* ws   = (float*)d_ws;
  float* x    = ws;                                         // N*H
  float* h    = x + (size_t)N_NODES * HIDDEN;               // N*H
  float* pool = h + (size_t)N_NODES * HIDDEN;               // G*3H
  float* cnt  = pool + (size_t)N_GRAPHS * N_LAYERS * HIDDEN;

  const int* src = eidx;
  const int* dst = eidx + N_EDGES;

  {
    int n = N_GRAPHS * N_LAYERS * HIDDEN + N_GRAPHS;
    zero_kernel<<<(n + 255) / 256, 256, 0, stream>>>(pool, n);
  }
  embed_kernel<<<(N_NODES * 32 + 255) / 256, 256, 0, stream>>>(z, ztab, x);

  for (int l = 0; l < N_LAYERS; ++l) {
    int n4 = N_NODES * HIDDEN / 4;
    copy_kernel<<<(n4 + 255) / 256, 256, 0, stream>>>((const float4*)x,
                                                      (float4*)h, n4);
    scatter_kernel<<<(N_EDGES * 32 + 255) / 256, 256, 0, stream>>>(x, src, dst, h);
    mlp_kernel<<<(N_NODES + MTILE - 1) / MTILE, TB, 0, stream>>>(
        h, W1 + l * HIDDEN * HIDDEN, b1 + l * HIDDEN,
        W2 + l * HIDDEN * HIDDEN, b2 + l * HIDDEN,
        lng + l * HIDDEN, lnb + l * HIDDEN, x);
    pool_kernel<<<(N_NODES * HIDDEN + 255) / 256, 256, 0, stream>>>(x, batch,
                                                                    pool, l);
  }
  cnt_kernel<<<(N_NODES + 255) / 256, 256, 0, stream>>>(batch, cnt);
  head_kernel<<<N_GRAPHS, 128, 0, stream>>>(pool, cnt, l1W, l1b, l2W, l2b, out);
}